// feat_gan_21388937134200
// MI455X (gfx1250) — compile-verified
//
#include <hip/hip_runtime.h>

// ---------------------------------------------------------------------------
// Problem constants (from reference setup_inputs)
// ---------------------------------------------------------------------------
#define B_   2
#define N_   4096      // source points per batch
#define S_   4096      // query points per batch (queries are always bat_xyz)
#define C_   128       // feature channels
#define GC_  131       // 3 xyz + 128 feat
#define NSMP 3
#define R2   1.0f      // radius^2

// workspace layout in 4-byte words
#define WS_ATT_IDX 0
#define WS_BAT_IDX (B_*S_*3)
#define WS_MASK    (2*B_*S_*3)
#define WS_PART    (2*B_*S_*3 + B_*S_)
#define NBLK_LOSS  96          // = B_*S_*NSMP / 256 triples per thread-block

typedef __attribute__((ext_vector_type(2))) float v2f;
typedef __attribute__((ext_vector_type(8))) float v8f;

// ---------------------------------------------------------------------------
// Kernel 1: WMMA ball query.
// One wave32 handles a tile of 16 query points; loops over 256 tiles of 16
// source points. Each 16x16 squared-distance tile is a single
// V_WMMA_F32_16X16X4_F32:  D = A(16x4) x B(4x16) + C,
//   A row m  = (qx, qy, qz, 1)
//   B col n  = (-2px, -2py, -2pz, |p|^2)
//   C[m][n]  = |q_m|^2          =>  D[m][n] = dist^2(q_m, p_n)
// C/D layout: VGPR r -> row r (lanes 0-15) and row r+8 (lanes 16-31),
// column == lane%16 == ballot bit position, so ballot(d<=R2) gives per-row
// "within" masks in ascending-column bit order. Owner lanes 0-15 extract the
// first 3 set bits across tiles (tiles scanned ascending) == jnp.sort + [:3].
//
// Latency plan: the whole 48KB per-batch point cloud is staged into LDS once
// per block (CDNA5 WGP has 320KB LDS), so the 256-iteration inner loop reads
// its B-fragment with a short-latency ds_load instead of stalling on L2 every
// iteration, and global traffic is amortized over the block's 4 waves.
//
// Fast path: with radius 1 in a 20^3 cube, >99% of 16x16 tiles contain zero
// in-radius pairs, so one min-reduce + one ballot gates the whole extraction.
// ---------------------------------------------------------------------------
__global__ __launch_bounds__(128)
void ball_query_wmma(const float* __restrict__ qxyz,   // [B,S,3] query points
                     const float* __restrict__ pxyz,   // [B,N,3] source points
                     int*  __restrict__ idx_out,       // [B,S,3]
                     int*  __restrict__ mask_out)      // [B,S] (may be null)
{
    __shared__ float shp[N_ * 3];                 // 48KB staged point cloud

    const int lane = threadIdx.x & 31;
    const int half = lane >> 4;          // 0: lanes 0-15, 1: lanes 16-31
    const int l    = lane & 15;
    const int wave = threadIdx.x >> 5;
    const int tile = blockIdx.x * 4 + wave;   // 16 queries per wave-tile
    const int b    = (blockIdx.x * 4) / (S_ / 16);   // uniform per block
    const int qb   = (tile % (S_ / 16)) * 16;

    const float* q = qxyz + (size_t)b * S_ * 3;
    const float* p = pxyz + (size_t)b * N_ * 3;

    // ---- stage the batch's point cloud into LDS (b128 in, b128 out) ------
    {
        const float4* gp4 = (const float4*)p;
        float4*       sp4 = (float4*)shp;
#pragma unroll
        for (int i = 0; i < (N_ * 3) / 4 / 128; ++i)      // 24 iterations
            sp4[i * 128 + threadIdx.x] = gp4[i * 128 + threadIdx.x];
    }

    // A fragment (16x4 f32): lanes 0-15 hold K={0,1}, lanes 16-31 K={2,3}
    const float qx = q[(qb + l) * 3 + 0];
    const float qy = q[(qb + l) * 3 + 1];
    const float qz = q[(qb + l) * 3 + 2];
    v2f a;
    a.x = half ? qz   : qx;
    a.y = half ? 1.0f : qy;

    // C fragment: c[r] = |q_{qb + r + 8*half}|^2  (matches D layout)
    float qn[8];
#pragma unroll
    for (int r = 0; r < 8; ++r) {
        const int m = qb + r + 8 * half;
        const float x = q[m * 3 + 0], y = q[m * 3 + 1], z = q[m * 3 + 2];
        qn[r] = x * x + y * y + z * z;
    }

    __syncthreads();                                  // staging complete

    int i0 = 0, i1 = 0, i2 = 0, cnt = 0;

    for (int t = 0; t < N_ / 16; ++t) {
        const int n0 = t * 16;
        const float* prow = &shp[(n0 + l) * 3];       // ds_load_b96
        const float px = prow[0], py = prow[1], pz = prow[2];
        const float pn = px * px + py * py + pz * pz;
        // B fragment (4x16 f32): lanes 0-15 rows K={0,1}, lanes 16-31 K={2,3}
        v2f bf;
        bf.x = half ? (-2.0f * pz) : (-2.0f * px);
        bf.y = half ? pn           : (-2.0f * py);

        v8f c;
#pragma unroll
        for (int r = 0; r < 8; ++r) c[r] = qn[r];

        // (neg_a, A, neg_b, B, c_mod, C, reuse_a, reuse_b)
        v8f d = __builtin_amdgcn_wmma_f32_16x16x4_f32(
            false, a, false, bf, (short)0, c, false, false);

        // Tile-level gate: skip extraction when no pair in this 16x16 tile is
        // within radius (the overwhelmingly common case). Uniform branch, so
        // EXEC is all-ones at the WMMA above on every iteration.
        float dmin = fminf(fminf(fminf(d[0], d[1]), fminf(d[2], d[3])),
                           fminf(fminf(d[4], d[5]), fminf(d[6], d[7])));
        const unsigned anyhit = __builtin_amdgcn_ballot_w32(dmin <= R2);
        if (anyhit != 0u) {
#pragma unroll
            for (int r = 0; r < 8; ++r) {
                const unsigned msk = __builtin_amdgcn_ballot_w32(d[r] <= R2);
                // Owner lane o (<16) owns row o; ballot r carries rows r, r+8.
                unsigned bits = 0u;
                if ((half == 0) && ((lane & 7) == r))
                    bits = (lane < 8) ? (msk & 0xFFFFu) : (msk >> 16);
                // Append up to 3 lowest set bits (ascending column order),
                // branchless so the wave reconverges before the next WMMA.
#pragma unroll
                for (int it = 0; it < 3; ++it) {
                    const bool take = (bits != 0u) && (cnt < NSMP);
                    const int  pos  = __builtin_ctz(bits | 0x10000u);
                    const int  nn   = n0 + pos;
                    i0 = (take && cnt == 0) ? nn : i0;
                    i1 = (take && cnt == 1) ? nn : i1;
                    i2 = (take && cnt == 2) ? nn : i2;
                    cnt += take ? 1 : 0;
                    bits = take ? (bits & (bits - 1u)) : bits;
                }
            }
            // cnt can only grow when a tile has hits -> check early-out here.
            const unsigned done =
                __builtin_amdgcn_ballot_w32((lane >= 16) || (cnt >= NSMP));
            if (done == 0xFFFFFFFFu) break;
        }
    }

    if (lane < 16) {
        const int s   = qb + lane;
        const int has = (cnt > 0) ? 1 : 0;
        const int f   = has ? i0 : 0;                 // pad with first valid, else 0
        const int j1  = (cnt > 1) ? i1 : f;
        const int j2  = (cnt > 2) ? i2 : f;
        const int base = (b * S_ + s) * 3;
        idx_out[base + 0] = f;
        idx_out[base + 1] = j1;
        idx_out[base + 2] = j2;
        if (mask_out) mask_out[b * S_ + s] = has;
    }
}

// ---------------------------------------------------------------------------
// Kernel 2: gather + masked squared-difference partial sums.
// One thread per (b,s,k) triple (24576 = 96 blocks x 256 threads exactly):
// mask/ai/bi loaded once, then a 131-channel loop. Everything is L2-resident
// (~8.6MB vs 192MB L2). Deterministic tree reduction (no float atomics).
// ---------------------------------------------------------------------------
__global__ __launch_bounds__(256)
void loss_partial(const float* __restrict__ att_xyz,
                  const float* __restrict__ bat_xyz,
                  const float* __restrict__ att_feat,   // [B,C,N]
                  const float* __restrict__ bat_feat,   // [B,C,N]
                  const int*   __restrict__ att_idx,    // [B,S,3]
                  const int*   __restrict__ bat_idx,    // [B,S,3]
                  const int*   __restrict__ mask,       // [B,S]
                  float*       __restrict__ partial)
{
    const int t  = blockIdx.x * blockDim.x + threadIdx.x;  // 0 .. B*S*NSMP-1
    // s fastest for coalesced mask/idx loads: t = s + S*(k + NSMP*b)
    const int s  = t % S_;
    const int k  = (t / S_) % NSMP;
    const int b  = t / (S_ * NSMP);
    const int bs = b * S_ + s;

    float acc = 0.0f;
    const int m  = mask[bs];
    if (m) {
        const int ai = att_idx[bs * 3 + k];
        const int bi = bat_idx[bs * 3 + k];
        const float* axyz = att_xyz + ((size_t)b * N_ + ai) * 3;
        const float* bxyz = bat_xyz + ((size_t)b * N_ + bi) * 3;
#pragma unroll
        for (int c = 0; c < 3; ++c) {
            const float dd = axyz[c] - bxyz[c];
            acc += dd * dd;
        }
        const float* af = att_feat + (size_t)b * C_ * N_ + ai;
        const float* bf = bat_feat + (size_t)b * C_ * N_ + bi;
        for (int c = 0; c < C_; ++c) {
            const float dd = af[(size_t)c * N_] - bf[(size_t)c * N_];
            acc += dd * dd;
        }
    }

    __shared__ float red[256];
    red[threadIdx.x] = acc;
    __syncthreads();
    for (int s2 = 128; s2 > 0; s2 >>= 1) {
        if (threadIdx.x < s2) red[threadIdx.x] += red[threadIdx.x + s2];
        __syncthreads();
    }
    if (threadIdx.x == 0) partial[blockIdx.x] = red[0];
}

__global__ __launch_bounds__(256)
void final_reduce(const float* __restrict__ partial, float* __restrict__ out)
{
    __shared__ float red[256];
    float acc = 0.0f;
    for (int i = threadIdx.x; i < NBLK_LOSS; i += 256) acc += partial[i];
    red[threadIdx.x] = acc;
    __syncthreads();
    for (int s2 = 128; s2 > 0; s2 >>= 1) {
        if (threadIdx.x < s2) red[threadIdx.x] += red[threadIdx.x + s2];
        __syncthreads();
    }
    if (threadIdx.x == 0) {
        const float inv = 1.0f / (float)((long long)B_ * S_ * NSMP * GC_);
        out[0] = red[0] * inv;   // mean; NaN guard in reference is a no-op
    }
}

// ---------------------------------------------------------------------------
// Launch: inputs are att_xyz, bat_xyz, att_feat_cn, bat_feat_cn (all f32).
// Queries are bat_xyz for BOTH ball queries (reference semantics).
// ---------------------------------------------------------------------------
extern "C" void kernel_launch(void* const* d_in, const int* in_sizes, int n_in,
                              void* d_out, int out_size, void* d_ws, size_t ws_size,
                              hipStream_t stream)
{
    const float* att_xyz  = (const float*)d_in[0];
    const float* bat_xyz  = (const float*)d_in[1];
    const float* att_feat = (const float*)d_in[2];
    const float* bat_feat = (const float*)d_in[3];

    int*   ws     = (int*)d_ws;
    int*   attIdx = ws + WS_ATT_IDX;
    int*   batIdx = ws + WS_BAT_IDX;
    int*   msk    = ws + WS_MASK;
    float* part   = (float*)(ws + WS_PART);

    const dim3 bq_grid(B_ * S_ / 64);   // 4 wave-tiles (16 queries each) / block
    const dim3 bq_block(128);

    // att_idx, mask = ball_point(att_xyz, bat_xyz)
    ball_query_wmma<<<bq_grid, bq_block, 0, stream>>>(bat_xyz, att_xyz, attIdx, msk);
    // bat_idx, _    = ball_point(bat_xyz, bat_xyz)
    ball_query_wmma<<<bq_grid, bq_block, 0, stream>>>(bat_xyz, bat_xyz, batIdx, nullptr);

    loss_partial<<<NBLK_LOSS, 256, 0, stream>>>(att_xyz, bat_xyz, att_feat, bat_feat,
                                                attIdx, batIdx, msk, part);
    final_reduce<<<1, 256, 0, stream>>>(part, (float*)d_out);
}